// Mix_Loss_30331059044854
// MI455X (gfx1250) — compile-verified
//
#include <hip/hip_runtime.h>

// ---------------------------------------------------------------------------
// Mix_Loss for MI455X (gfx1250, wave32).
// Memory-bound streaming reduction: ~404 MB read -> 2 scalars.
// loss2 swap is a no-op for the forward value ((z1-z2)^2 symmetric), so we
// stream z once with nontemporal B128 loads. Wave reduction uses
// V_WMMA_F32_16X16X4_F32 (A = lane partials in K0/K2, B = ones) which is an
// exact 2:1 fold across the wave halves, finished with one shfl_xor(16).
// Deterministic two-pass reduction (no float atomics).
// ---------------------------------------------------------------------------

typedef __attribute__((ext_vector_type(2))) float v2f;
typedef __attribute__((ext_vector_type(4))) float v4f;
typedef __attribute__((ext_vector_type(8))) float v8f;

#define NBLK 1024
#define NTHR 256
#define NWAVE (NTHR / 32)

// Exact sum of 32 lane partials via one WMMA.
// A (16x4 f32, 2 VGPRs/lane): VGPR0 = p (lanes0-15 -> A[m][0], lanes16-31 ->
// A[m][2]), VGPR1 = 0. B = all ones => D[m][n] = p_m + p_{m+16} for every n.
// Lane l<16 holds D[0..7][l] in its 8 acc VGPRs, lane l+16 holds D[8..15][l];
// summing the 8 VGPRs and adding the cross-half partner gives the full sum.
__device__ __forceinline__ float wave_sum32(float p) {
  v2f a;
  a.x = p;
  a.y = 0.0f;
  v2f b;
  b.x = 1.0f;
  b.y = 1.0f;
  v8f c = {0.f, 0.f, 0.f, 0.f, 0.f, 0.f, 0.f, 0.f};
  c = __builtin_amdgcn_wmma_f32_16x16x4_f32(false, a, false, b, (short)0, c,
                                            false, false);
  float s = ((c[0] + c[1]) + (c[2] + c[3])) + ((c[4] + c[5]) + (c[6] + c[7]));
  s += __shfl_xor(s, 16, 32);
  return s;  // full 32-lane sum, uniform across the wave
}

__device__ __forceinline__ float bce_logit(float x, float y) {
  // max(x,0) - x*y + log1p(exp(-|x|))
  return fmaxf(x, 0.0f) - x * y + log1pf(expf(-fabsf(x)));
}

__global__ __launch_bounds__(NTHR) void mix_loss_main(
    const float* __restrict__ z, const float* __restrict__ pred,
    const float* __restrict__ tgt, const float* __restrict__ wts,
    const int* __restrict__ dividep, float* __restrict__ ws, int B) {
  const int tid = blockIdx.x * NTHR + threadIdx.x;
  const int stride = NBLK * NTHR;

  // ---- loss2 partial: sum of (z[2i]-z[2i+1])^2, streamed as float4 ----
  const v4f* __restrict__ z4 = (const v4f*)z;
  const int NZ4 = B * 96;  // 96 float4 per 384-float row
  float acc2 = 0.0f;
  const int kz = NZ4 / stride;  // exactly 48 for the reference shapes
#pragma unroll 4
  for (int it = 0; it < kz; ++it) {
    const int k = tid + it * stride;
    const int i = k / 96;
    const int j = k - i * 96;
    const int base = i * 192 + j;  // row 2i, chunk j
    v4f a = __builtin_nontemporal_load(z4 + base);
    v4f b = __builtin_nontemporal_load(z4 + base + 96);  // row 2i+1
    const float dx = a.x - b.x, dy = a.y - b.y;
    const float dz = a.z - b.z, dw = a.w - b.w;
    acc2 += dx * dx + dy * dy + dz * dz + dw * dw;
  }
  {  // remainder (zero for the reference shapes, kept for generality)
    const int k = tid + kz * stride;
    if (k < NZ4) {
      const int i = k / 96;
      const int j = k - i * 96;
      const int base = i * 192 + j;
      v4f a = __builtin_nontemporal_load(z4 + base);
      v4f b = __builtin_nontemporal_load(z4 + base + 96);
      const float dx = a.x - b.x, dy = a.y - b.y;
      const float dz = a.z - b.z, dw = a.w - b.w;
      acc2 += dx * dx + dy * dy + dz * dz + dw * dw;
    }
  }

  // ---- loss1 partial: weighted MSE/BCE over pred pairs ----
  const int dv = *dividep;  // uniform scalar load
  float w[8];
#pragma unroll
  for (int j = 0; j < 8; ++j) w[j] = wts[j];

  const v4f* __restrict__ p4 = (const v4f*)pred;
  const v4f* __restrict__ t4 = (const v4f*)tgt;
  float acc1 = 0.0f;
  for (int i = tid; i < B; i += stride) {
    v4f p0a = p4[4 * i + 0], p0b = p4[4 * i + 1];  // pred row 2i
    v4f p1a = p4[4 * i + 2], p1b = p4[4 * i + 3];  // pred row 2i+1
    v4f ta = t4[2 * i + 0], tb = t4[2 * i + 1];    // target row i
    const float x0[8] = {p0a.x, p0a.y, p0a.z, p0a.w, p0b.x, p0b.y, p0b.z, p0b.w};
    const float x1[8] = {p1a.x, p1a.y, p1a.z, p1a.w, p1b.x, p1b.y, p1b.z, p1b.w};
    const float tt[8] = {ta.x, ta.y, ta.z, ta.w, tb.x, tb.y, tb.z, tb.w};
#pragma unroll
    for (int j = 0; j < 8; ++j) {
      float f0, f1;
      if (j < dv) {  // dv is uniform -> non-divergent branch
        const float d0 = x0[j] - tt[j];
        const float d1 = x1[j] - tt[j];
        f0 = d0 * d0;
        f1 = d1 * d1;
      } else {
        f0 = bce_logit(x0[j], tt[j]);
        f1 = bce_logit(x1[j], tt[j]);
      }
      acc1 += w[j] * (f0 + f1);
    }
  }

  // ---- block reduction (WMMA wave sums, EXEC all-ones here) ----
  const float s1 = wave_sum32(acc1);
  const float s2 = wave_sum32(acc2);

  __shared__ float sh1[NWAVE];
  __shared__ float sh2[NWAVE];
  const int wv = threadIdx.x >> 5;
  const int ln = threadIdx.x & 31;
  if (ln == 0) {
    sh1[wv] = s1;
    sh2[wv] = s2;
  }
  __syncthreads();
  if (threadIdx.x == 0) {
    float t1 = 0.0f, t2 = 0.0f;
#pragma unroll
    for (int k = 0; k < NWAVE; ++k) {
      t1 += sh1[k];
      t2 += sh2[k];
    }
    ws[blockIdx.x] = t1;         // loss1 partial
    ws[NBLK + blockIdx.x] = t2;  // loss2 partial
  }
}

__global__ __launch_bounds__(NTHR) void mix_loss_final(
    const float* __restrict__ ws, float* __restrict__ out, int B) {
  float a1 = 0.0f, a2 = 0.0f;
  for (int i = threadIdx.x; i < NBLK; i += NTHR) {
    a1 += ws[i];
    a2 += ws[NBLK + i];
  }
  const float s1 = wave_sum32(a1);
  const float s2 = wave_sum32(a2);

  __shared__ float sh1[NWAVE];
  __shared__ float sh2[NWAVE];
  const int wv = threadIdx.x >> 5;
  const int ln = threadIdx.x & 31;
  if (ln == 0) {
    sh1[wv] = s1;
    sh2[wv] = s2;
  }
  __syncthreads();
  if (threadIdx.x == 0) {
    float t1 = 0.0f, t2 = 0.0f;
#pragma unroll
    for (int k = 0; k < NWAVE; ++k) {
      t1 += sh1[k];
      t2 += sh2[k];
    }
    out[0] = t1 / (float)B;                                   // loss1
    out[1] = (float)((double)t2 / ((double)B * 384.0));       // loss2
  }
}

extern "C" void kernel_launch(void* const* d_in, const int* in_sizes, int n_in,
                              void* d_out, int out_size, void* d_ws,
                              size_t ws_size, hipStream_t stream) {
  const float* z = (const float*)d_in[0];     // (2B, 384) f32
  const float* pred = (const float*)d_in[1];  // (2B, 8)   f32
  const float* tgt = (const float*)d_in[2];   // (B, 8)    f32
  const float* wts = (const float*)d_in[3];   // (8,)      f32
  const int* divp = (const int*)d_in[4];      // scalar    i32
  const int B = in_sizes[2] / 8;

  float* ws = (float*)d_ws;  // 2*NBLK floats of partials (fully overwritten)
  mix_loss_main<<<NBLK, NTHR, 0, stream>>>(z, pred, tgt, wts, divp, ws, B);
  mix_loss_final<<<1, NTHR, 0, stream>>>(ws, (float*)d_out, B);
}